// MultiLinear_14224931684419
// MI455X (gfx1250) — compile-verified
//
#include <hip/hip_runtime.h>
#include <hip/hip_bf16.h>

typedef __attribute__((ext_vector_type(2))) float v2f;
typedef __attribute__((ext_vector_type(8))) float v8f;

#define NUM_LAYERS 64
#define IN_F 128
#define OUT_F 128
#define SPLITS 8

// ---------------- grouping kernels ----------------

__global__ void ml_zero_counts(int* counts) {
    counts[threadIdx.x] = 0;
}

__global__ void ml_histogram(const int* __restrict__ idx, int* __restrict__ counts, int batch) {
    int b = blockIdx.x * blockDim.x + threadIdx.x;
    if (b < batch) atomicAdd(&counts[idx[b] & (NUM_LAYERS - 1)], 1);
}

__global__ void ml_scan(const int* __restrict__ counts, int* __restrict__ offsets,
                        int* __restrict__ cursors) {
    if (threadIdx.x == 0) {
        int acc = 0;
        for (int i = 0; i < NUM_LAYERS; ++i) { offsets[i] = acc; acc += counts[i]; }
    }
    __syncthreads();
    cursors[threadIdx.x] = offsets[threadIdx.x];
}

__global__ void ml_scatter(const int* __restrict__ idx, int* __restrict__ cursors,
                           int* __restrict__ perm, int batch) {
    int b = blockIdx.x * blockDim.x + threadIdx.x;
    if (b < batch) {
        int l = idx[b] & (NUM_LAYERS - 1);
        int pos = atomicAdd(&cursors[l], 1);
        perm[pos] = b;
    }
}

// ---------------- WMMA GEMM: out[16,128] = X[16,128] @ W[128,128] + bias ----------------
// grid = (NUM_LAYERS, SPLITS), block = 256 (8 waves; wave w owns output columns [16w,16w+16))

__global__ __launch_bounds__(256)
void ml_wmma_gemm(const float* __restrict__ x, const float* __restrict__ w,
                  const float* __restrict__ bias,
                  const int* __restrict__ offsets, const int* __restrict__ counts,
                  const int* __restrict__ perm, float* __restrict__ out) {
    const int layer = blockIdx.x;
    const int split = blockIdx.y;
    const int tid   = threadIdx.x;
    const int lane  = tid & 31;
    const int wave  = tid >> 5;       // n-strip index 0..7
    const int g     = lane >> 4;      // half-group of the wave
    const int ln    = lane & 15;
    const int n0    = wave * 16;

    // padded stride 132 (== 4 mod 64 banks) so column reads of A frags are conflict-free
    __shared__ float xs[16 * 132];

    const float* wl = w + (size_t)layer * IN_F * OUT_F;

    // Preload B fragments once per block: lane holds W[k][n0+ln] for k = 4t + 2g + {0,1}
    v2f bfrag[32];
#pragma unroll
    for (int t = 0; t < 32; ++t) {
        int k = 4 * t + 2 * g;
        bfrag[t].x = wl[(size_t)k * OUT_F + n0 + ln];
        bfrag[t].y = wl[(size_t)(k + 1) * OUT_F + n0 + ln];
    }

    const float bv = bias[layer * OUT_F + n0 + ln];

    const int base   = offsets[layer];
    const int cnt    = counts[layer];
    const int ntiles = (cnt + 15) >> 4;

    for (int t = split; t < ntiles; t += SPLITS) {
        const int row0  = t * 16;
        const int nrows = min(16, cnt - row0);

        // stage 16 gathered rows into LDS, coalesced float4, zero-fill the ragged tail
        for (int j = tid; j < 16 * 32; j += 256) {
            int r  = j >> 5;   // row 0..15
            int c4 = j & 31;   // float4 column 0..31
            float4 v = make_float4(0.f, 0.f, 0.f, 0.f);
            if (r < nrows) {
                int src = perm[base + row0 + r];
                v = ((const float4*)(x + (size_t)src * IN_F))[c4];
            }
            *(float4*)(&xs[r * 132 + c4 * 4]) = v;
        }
        __syncthreads();

        // K = 128 as 32 chained v_wmma_f32_16x16x4_f32
        v8f acc = {};
#pragma unroll
        for (int kt = 0; kt < 32; ++kt) {
            int k = 4 * kt + 2 * g;
            v2f a = *(const v2f*)(&xs[ln * 132 + k]);   // A[m=ln][k], A[m=ln][k+1]
            acc = __builtin_amdgcn_wmma_f32_16x16x4_f32(
                /*neg_a=*/false, a, /*neg_b=*/false, bfrag[kt],
                /*c_mod=*/(short)0, acc, /*reuse_a=*/false, /*reuse_b=*/false);
        }

        // bias + scatter-store: C/D row M = r + 8g, col n0+ln
#pragma unroll
        for (int r = 0; r < 8; ++r) {
            int M = r + 8 * g;
            if (M < nrows) {
                int dst = perm[base + row0 + M];
                out[(size_t)dst * OUT_F + n0 + ln] = acc[r] + bv;
            }
        }
        __syncthreads();   // xs reused next tile
    }
}

// ---------------- launch ----------------

extern "C" void kernel_launch(void* const* d_in, const int* in_sizes, int n_in,
                              void* d_out, int out_size, void* d_ws, size_t ws_size,
                              hipStream_t stream) {
    const float* x    = (const float*)d_in[0];
    const int*   idx  = (const int*)d_in[1];
    const float* w    = (const float*)d_in[2];
    const float* bias = (const float*)d_in[3];
    float*       out  = (float*)d_out;

    const int batch = in_sizes[0] / IN_F;

    // workspace layout: counts[64] | offsets[64] | cursors[64] | perm[batch]
    int* wsI     = (int*)d_ws;
    int* counts  = wsI;
    int* offsets = wsI + 64;
    int* cursors = wsI + 128;
    int* perm    = wsI + 192;

    const int blocks = (batch + 255) / 256;

    ml_zero_counts<<<1, NUM_LAYERS, 0, stream>>>(counts);
    ml_histogram <<<blocks, 256, 0, stream>>>(idx, counts, batch);
    ml_scan      <<<1, NUM_LAYERS, 0, stream>>>(counts, offsets, cursors);
    ml_scatter   <<<blocks, 256, 0, stream>>>(idx, cursors, perm, batch);
    ml_wmma_gemm <<<dim3(NUM_LAYERS, SPLITS), 256, 0, stream>>>(
        x, w, bias, offsets, counts, perm, out);
}